// CompLayer_1082331759043
// MI455X (gfx1250) — compile-verified
//
#include <hip/hip_runtime.h>
#include <math.h>

// ---------------- problem constants ----------------
#define N_ENT  1968
#define NREL2  116          // 2*N_REL
#define EDGES  20000
#define S_TOK  16
#define HDIM   128
#define BQ     8
#define TQ     20
#define KSEL   196          // int(N_ENT*0.1)

typedef __attribute__((ext_vector_type(2))) float v2f;
typedef __attribute__((ext_vector_type(8))) float v8f;

// V_WMMA_F32_16X16X4_F32 : D(16x16 f32) = A(16x4 f32) * B(4x16 f32) + C
static __device__ __forceinline__ v8f wmma4(v2f a, v2f b, v8f c) {
  return __builtin_amdgcn_wmma_f32_16x16x4_f32(false, a, false, b, (short)0, c,
                                               false, false);
}

// ---- CDNA5 async global->LDS copy path (ASYNCcnt), with portable fallback --
// ROCm 7.2 signature (from diagnostics): (int4* gsrc, int4* lds, imm, imm)
#if defined(__AMDGCN__)
#if __has_builtin(__builtin_amdgcn_global_load_async_to_lds_b128) && \
    __has_builtin(__builtin_amdgcn_s_wait_asynccnt)
#define HAVE_ASYNC_LDS 1
typedef int v4i_vs __attribute__((__vector_size__(16)));
#endif
#endif
#ifndef HAVE_ASYNC_LDS
#define HAVE_ASYNC_LDS 0
#endif

// ---------------------------------------------------------------------------
// fp32 WMMA GEMM with LDS-staged B:  C[M,128] = A[M,K] @ B[K,128] (+ bias)
// grid.x = M/16 (one 16-row stripe per block); block = 256 = 8 waves,
// wave w computes the 16x16 tile at columns [16w, 16w+16).
// B is staged chunk-by-chunk (32x128 = 16KB) into LDS via async-to-LDS copies
// (one 16B transfer per lane, ASYNCcnt tracked), then fragments are read from
// LDS.  K % 32 == 0.
// Fragment layout (ISA 7.12.2, 32-bit):
//   A: lane<16 holds row lm, K = k,k+1 ; lane>=16 holds row lm, K = k+2,k+3
//   B: lane<16 holds col lm, K = k,k+1 ; lane>=16 holds col lm, K = k+2,k+3
//   C: VGPR r -> row r (lanes 0-15) / row r+8 (lanes 16-31), col = lm
// ---------------------------------------------------------------------------
__global__ void gemm_wmma_f32(const float* __restrict__ A,
                              const float* __restrict__ Bm,
                              const float* __restrict__ bias,
                              float* __restrict__ C, int M, int K) {
  const int wave = threadIdx.x >> 5;
  const int lane = threadIdx.x & 31;
  const int half = lane >> 4, lm = lane & 15;
  const int m0 = blockIdx.x << 4;
  const int n0 = wave << 4;
  __shared__ __align__(16) float Bs[32 * 128];   // 16KB k-chunk of B
  const float* Arow = A + (size_t)(m0 + lm) * K;
  v8f acc = {0.f, 0.f, 0.f, 0.f, 0.f, 0.f, 0.f, 0.f};
  for (int k0 = 0; k0 < K; k0 += 32) {
    // ---- stage B[k0..k0+32) x [0..128) (contiguous 16KB) into LDS ----
    const float* gsrc = Bm + (size_t)k0 * 128;
    for (int i = threadIdx.x; i < 1024; i += 256) {    // 1024 x 16B
#if HAVE_ASYNC_LDS
      __builtin_amdgcn_global_load_async_to_lds_b128(
          (v4i_vs*)(gsrc + i * 4), (v4i_vs*)(Bs + i * 4), 0, 0);
#else
      ((float4*)Bs)[i] = ((const float4*)gsrc)[i];
#endif
    }
#if HAVE_ASYNC_LDS
    __builtin_amdgcn_s_wait_asynccnt(0);
#endif
    __syncthreads();
    // ---- 8 WMMA k-steps out of LDS ----
    for (int k = 0; k < 32; k += 4) {
      const int ka = k + 2 * half;
      v2f a, b;
      a.x = Arow[k0 + ka];
      a.y = Arow[k0 + ka + 1];
      b.x = Bs[ka * 128 + n0 + lm];
      b.y = Bs[(ka + 1) * 128 + n0 + lm];
      acc = wmma4(a, b, acc);
    }
    __syncthreads();
  }
  const float bv = bias ? bias[n0 + lm] : 0.f;
  for (int r = 0; r < 8; ++r)
    C[(size_t)(m0 + r + 8 * half) * 128 + n0 + lm] = acc[r] + bv;
}

// ------------- qsum[b,h] = sum_t (question_emb[b,t,:]@W_tok + b_tok)[h] ----
__global__ void qsum_kernel(const float* __restrict__ qe,
                            const float* __restrict__ Wtok,
                            const float* __restrict__ btok,
                            float* __restrict__ qsum) {
  const int tid = blockIdx.x * blockDim.x + threadIdx.x;   // 1024
  if (tid >= BQ * HDIM) return;
  const int b = tid >> 7, h = tid & 127;
  float acc = (float)TQ * btok[h];
  for (int d = 0; d < 2 * HDIM; ++d) {
    float qs = 0.f;
    for (int t = 0; t < TQ; ++t) qs += qe[((size_t)(b * TQ + t)) * (2 * HDIM) + d];
    acc += qs * Wtok[(size_t)d * HDIM + h];
  }
  qsum[b * HDIM + h] = acc;
}

// ------------- eps[rel] = max(1, #nonzero mask) ----------------------------
__global__ void eps_kernel(const float* __restrict__ rel_mask,
                           float* __restrict__ epsv) {
  const int r = threadIdx.x;
  if (r >= NREL2) return;
  float cnt = 0.f;
  for (int s = 0; s < S_TOK; ++s) cnt += (rel_mask[r * S_TOK + s] != 0.f) ? 1.f : 0.f;
  epsv[r] = (cnt == 0.f) ? 1.f : cnt;
}

// ------------- per-edge: comp_feat[e,h] and raw attention w[b,e] -----------
__global__ void edge_kernel(const float* __restrict__ Akey,
                            const float* __restrict__ Rkey,
                            const float* __restrict__ qsum,
                            const float* __restrict__ epsv,
                            const float* __restrict__ Wcomp,
                            const float* __restrict__ bcomp,
                            const int* __restrict__ src,
                            const int* __restrict__ dst,
                            const int* __restrict__ rel,
                            const float* __restrict__ rel_mask,
                            float* __restrict__ comp_feat,
                            float* __restrict__ wbe) {
  const int e = blockIdx.x, h = threadIdx.x;   // 128 threads
  const int si = src[e], di = dst[e], ri = rel[e];
  const float* As = Akey + (size_t)si * S_TOK * HDIM + h;
  const float* Ad = Akey + (size_t)di * S_TOK * HDIM + h;
  const float* Rr = Rkey + (size_t)ri * S_TOK * HDIM + h;
  const float* mk = rel_mask + ri * S_TOK;
  float wv = 0.f, cf = 0.f;
  for (int s = 0; s < S_TOK; ++s) {
    const float ke = As[s * HDIM] + Rr[s * HDIM] - Ad[s * HDIM];
    wv += mk[s] * ke;
    cf += Wcomp[s] * ke;
  }
  comp_feat[(size_t)e * HDIM + h] = cf + bcomp[0];
  __shared__ float sh[HDIM];
  sh[h] = wv;
  __syncthreads();
  if (h < BQ) {
    const float* qb = qsum + h * HDIM;
    float acc = 0.f;
    for (int j = 0; j < HDIM; ++j) acc += qb[j] * sh[j];
    const float ce = 0.125f / (epsv[ri] * (float)TQ);   // scale=(H/2)^-0.5
    wbe[(size_t)h * EDGES + e] = ce * acc;
  }
}

// ------------- softmax over 20000 edges, per batch -------------------------
__global__ void softmax_kernel(const float* __restrict__ wbe,
                               float* __restrict__ atts) {
  const int b = blockIdx.x, tid = threadIdx.x;   // 256 threads
  __shared__ float red[256];
  const float* wr = wbe + (size_t)b * EDGES;
  float mx = -INFINITY;
  for (int i = tid; i < EDGES; i += 256) mx = fmaxf(mx, wr[i]);
  red[tid] = mx; __syncthreads();
  for (int s = 128; s > 0; s >>= 1) { if (tid < s) red[tid] = fmaxf(red[tid], red[tid + s]); __syncthreads(); }
  mx = red[0]; __syncthreads();
  float sum = 0.f;
  for (int i = tid; i < EDGES; i += 256) sum += expf(wr[i] - mx);
  red[tid] = sum; __syncthreads();
  for (int s = 128; s > 0; s >>= 1) { if (tid < s) red[tid] += red[tid + s]; __syncthreads(); }
  const float inv = 1.f / red[0];
  for (int i = tid; i < EDGES; i += 256)
    atts[(size_t)b * EDGES + i] = expf(wr[i] - mx) * inv;
}

// ------------- segment max #1 and #2 over src (atts > 0 -> uint order OK) --
__global__ void seg_m1(const float* __restrict__ atts,
                       const int* __restrict__ src, unsigned* __restrict__ m1) {
  const int idx = blockIdx.x * blockDim.x + threadIdx.x;
  if (idx >= BQ * EDGES) return;
  const int b = idx / EDGES, e = idx % EDGES;
  atomicMax(&m1[b * N_ENT + src[e]], __float_as_uint(atts[idx]));
}
__global__ void seg_m2(const float* __restrict__ atts,
                       const int* __restrict__ src,
                       const unsigned* __restrict__ m1,
                       unsigned* __restrict__ m2) {
  const int idx = blockIdx.x * blockDim.x + threadIdx.x;
  if (idx >= BQ * EDGES) return;
  const int b = idx / EDGES, e = idx % EDGES;
  const unsigned av = __float_as_uint(atts[idx]);
  if (av != m1[b * N_ENT + src[e]])
    atomicMax(&m2[b * N_ENT + src[e]], av);
}

// ------------- gated message scatter-add to neigh --------------------------
__global__ void msg_kernel(const float* __restrict__ atts,
                           const float* __restrict__ comp_feat,
                           const int* __restrict__ src,
                           const int* __restrict__ dst,
                           const unsigned* __restrict__ m2,
                           float* __restrict__ neigh) {
  const int e = blockIdx.x, h = threadIdx.x;
  const int si = src[e], di = dst[e];
  const float cfh = comp_feat[(size_t)e * HDIM + h];
  for (int b = 0; b < BQ; ++b) {
    const float a = atts[(size_t)b * EDGES + e];
    if (a >= __uint_as_float(m2[b * N_ENT + si]))
      atomicAdd(&neigh[((size_t)b * N_ENT + di) * HDIM + h], cfh * a);
  }
}

// ------------- LayerNorm over H ------------------------------------------
__global__ void ln_kernel(const float* __restrict__ neigh, float* __restrict__ x) {
  const int row = blockIdx.x, h = threadIdx.x;   // 128 threads
  __shared__ float red[HDIM];
  const float v = neigh[(size_t)row * HDIM + h];
  red[h] = v; __syncthreads();
  for (int s = 64; s > 0; s >>= 1) { if (h < s) red[h] += red[h + s]; __syncthreads(); }
  const float mu = red[0] / (float)HDIM;
  __syncthreads();
  const float d = v - mu;
  red[h] = d * d; __syncthreads();
  for (int s = 64; s > 0; s >>= 1) { if (h < s) red[h] += red[h + s]; __syncthreads(); }
  const float rstd = rsqrtf(red[0] / (float)HDIM + 1e-5f);
  x[(size_t)row * HDIM + h] = d * rstd;
}

// ------------- self-attention: streaming softmax, all GEMMs via WMMA -------
// One wave per (b, 16-query-row tile): 8*123 = 984 tasks, 123 blocks x 8 waves.
__global__ void attn_kernel(const float* __restrict__ Qm,
                            const float* __restrict__ Km,
                            const float* __restrict__ Vm,
                            float* __restrict__ attended,
                            float* __restrict__ importance) {
  const int wave = threadIdx.x >> 5, lane = threadIdx.x & 31;
  const int half = lane >> 4, lm = lane & 15;
  const int NT = N_ENT / 16;                 // 123 key tiles
  const int task = blockIdx.x * 8 + wave;    // exactly 984
  const int b = task / NT, rt = task % NT;
  const float* Qb = Qm + (size_t)b * N_ENT * HDIM;
  const float* Kb = Km + (size_t)b * N_ENT * HDIM;
  const float* Vb = Vm + (size_t)b * N_ENT * HDIM;
  __shared__ float P_[8][256];
  __shared__ float rmax_[8][16];
  __shared__ float rinv_[8][16];
  float* P = P_[wave];
  const float sqrtH = 11.313708498984761f;   // scores = QK^T / H^-0.5
  const float* Qrow = Qb + (size_t)(rt * 16 + lm) * HDIM;

  // ---- pass 1: row max + row sum (online) ----
  float m_run = -INFINITY, s_run = 0.f;
  for (int mt = 0; mt < NT; ++mt) {
    v8f c = {0.f, 0.f, 0.f, 0.f, 0.f, 0.f, 0.f, 0.f};
    const float* Krow = Kb + (size_t)(mt * 16 + lm) * HDIM;
    for (int k = 0; k < HDIM; k += 4) {
      const int ka = k + 2 * half;
      v2f a, bb;
      a.x = Qrow[ka];  a.y = Qrow[ka + 1];
      bb.x = Krow[ka]; bb.y = Krow[ka + 1];
      c = wmma4(a, bb, c);
    }
    for (int r = 0; r < 8; ++r) P[(r + 8 * half) * 16 + lm] = c[r] * sqrtH;
    __syncthreads();
    float tmax = -INFINITY;
    for (int j = 0; j < 16; ++j) tmax = fmaxf(tmax, P[lm * 16 + j]);
    const float m_new = fmaxf(m_run, tmax);
    float ts = 0.f;
    for (int j = 0; j < 16; ++j) ts += expf(P[lm * 16 + j] - m_new);
    s_run = s_run * expf(m_run - m_new) + ts;
    m_run = m_new;
    __syncthreads();
  }
  if (half == 0) { rmax_[wave][lm] = m_run; rinv_[wave][lm] = 1.f / s_run; }
  __syncthreads();

  // ---- pass 2: P = softmax tile; attended += P @ V ; importance += colsum(P)
  v8f acc[8];
  for (int f = 0; f < 8; ++f) acc[f] = (v8f){0.f, 0.f, 0.f, 0.f, 0.f, 0.f, 0.f, 0.f};
  for (int mt = 0; mt < NT; ++mt) {
    v8f c = {0.f, 0.f, 0.f, 0.f, 0.f, 0.f, 0.f, 0.f};
    const float* Krow = Kb + (size_t)(mt * 16 + lm) * HDIM;
    for (int k = 0; k < HDIM; k += 4) {
      const int ka = k + 2 * half;
      v2f a, bb;
      a.x = Qrow[ka];  a.y = Qrow[ka + 1];
      bb.x = Krow[ka]; bb.y = Krow[ka + 1];
      c = wmma4(a, bb, c);
    }
    for (int r = 0; r < 8; ++r) {
      const int row = r + 8 * half;
      P[row * 16 + lm] = expf(c[r] * sqrtH - rmax_[wave][row]) * rinv_[wave][row];
    }
    __syncthreads();
    if (half == 0) {   // importance: column sums of this P tile
      float ssum = 0.f;
      for (int r = 0; r < 16; ++r) ssum += P[r * 16 + lm];
      atomicAdd(&importance[b * N_ENT + mt * 16 + lm], ssum);
    }
    for (int f = 0; f < 8; ++f) {
      for (int kk = 0; kk < 16; kk += 4) {
        const int ka = kk + 2 * half;
        v2f a, bb;
        a.x = P[lm * 16 + ka]; a.y = P[lm * 16 + ka + 1];
        bb.x = Vb[(size_t)(mt * 16 + ka) * HDIM + f * 16 + lm];
        bb.y = Vb[(size_t)(mt * 16 + ka + 1) * HDIM + f * 16 + lm];
        acc[f] = wmma4(a, bb, acc[f]);
      }
    }
    __syncthreads();
  }
  for (int f = 0; f < 8; ++f)
    for (int r = 0; r < 8; ++r)
      attended[((size_t)b * N_ENT + rt * 16 + r + 8 * half) * HDIM + f * 16 + lm] = acc[f][r];
}

// ------------- iterative top-196 per batch (jax tie rule: lowest idx) ------
__global__ void topk_kernel(const float* __restrict__ importance,
                            int* __restrict__ selidx) {
  const int b = blockIdx.x, tid = threadIdx.x;   // 256 threads
  __shared__ float vals[N_ENT];
  __shared__ float rv[256];
  __shared__ int ri[256];
  for (int i = tid; i < N_ENT; i += 256) vals[i] = importance[b * N_ENT + i];
  __syncthreads();
  for (int k = 0; k < KSEL; ++k) {
    float best = -INFINITY; int bi = N_ENT;
    for (int i = tid; i < N_ENT; i += 256) {
      const float v = vals[i];
      if (v > best || (v == best && i < bi)) { best = v; bi = i; }
    }
    rv[tid] = best; ri[tid] = bi; __syncthreads();
    for (int s = 128; s > 0; s >>= 1) {
      if (tid < s) {
        if (rv[tid + s] > rv[tid] || (rv[tid + s] == rv[tid] && ri[tid + s] < ri[tid])) {
          rv[tid] = rv[tid + s]; ri[tid] = ri[tid + s];
        }
      }
      __syncthreads();
    }
    if (tid == 0) { selidx[b * KSEL + k] = ri[0]; vals[ri[0]] = -INFINITY; }
    __syncthreads();
  }
}

// ------------- out = tanh(attended[b, idx] @ W_kg + b_kg) via WMMA ---------
// M = B*KSEL = 1568 (98 tiles), N = 128 (8 tiles): 98 blocks x 8 waves.
__global__ void out_kernel(const float* __restrict__ attended,
                           const int* __restrict__ selidx,
                           const float* __restrict__ Wkg,
                           const float* __restrict__ bkg,
                           float* __restrict__ out) {
  const int wave = threadIdx.x >> 5, lane = threadIdx.x & 31;
  const int half = lane >> 4, lm = lane & 15;
  const int tile = blockIdx.x * 8 + wave;
  const int mt = tile >> 3, nt = tile & 7;
  const int m0 = mt << 4, n0 = nt << 4;
  const int row = m0 + lm;                    // < 1568
  const int b = row / KSEL, kk = row % KSEL;
  const float* Arow = attended + ((size_t)b * N_ENT + selidx[b * KSEL + kk]) * HDIM;
  v8f acc = {0.f, 0.f, 0.f, 0.f, 0.f, 0.f, 0.f, 0.f};
  for (int k = 0; k < HDIM; k += 4) {
    const int ka = k + 2 * half;
    v2f a, bb;
    a.x = Arow[ka]; a.y = Arow[ka + 1];
    bb.x = Wkg[(size_t)ka * HDIM + n0 + lm];
    bb.y = Wkg[(size_t)(ka + 1) * HDIM + n0 + lm];
    acc = wmma4(a, bb, acc);
  }
  const float bv = bkg[n0 + lm];
  for (int r = 0; r < 8; ++r)
    out[(size_t)(m0 + r + 8 * half) * HDIM + n0 + lm] = tanhf(acc[r] + bv);
}

// ---------------------------------------------------------------------------
extern "C" void kernel_launch(void* const* d_in, const int* in_sizes, int n_in,
                              void* d_out, int out_size, void* d_ws, size_t ws_size,
                              hipStream_t stream) {
  (void)in_sizes; (void)n_in; (void)out_size; (void)ws_size;
  // setup_inputs() dict order:
  const float* ent_emb = (const float*)d_in[0];   // (1968,16,256)
  const float* rel_emb = (const float*)d_in[1];   // (116,16,256)
  /* d_in[2] ent_mask unused by reference */
  const float* rel_mask = (const float*)d_in[3];  // (116,16)
  const float* q_emb   = (const float*)d_in[4];   // (8,20,256)
  const int*   src     = (const int*)d_in[5];
  const int*   dst     = (const int*)d_in[6];
  const int*   rel_id  = (const int*)d_in[7];
  const float* W_tok = (const float*)d_in[8];
  const float* b_tok = (const float*)d_in[9];
  const float* W_key = (const float*)d_in[10];
  const float* b_key = (const float*)d_in[11];
  const float* W_comp = (const float*)d_in[12];
  const float* b_comp = (const float*)d_in[13];
  const float* Wq = (const float*)d_in[14];
  const float* bq = (const float*)d_in[15];
  const float* Wk = (const float*)d_in[16];
  const float* bk = (const float*)d_in[17];
  const float* Wv = (const float*)d_in[18];
  const float* bv = (const float*)d_in[19];
  const float* W_kg = (const float*)d_in[20];
  const float* b_kg = (const float*)d_in[21];
  float* out = (float*)d_out;

  // ---- workspace layout (float offsets) ----
  float* ws = (float*)d_ws;
  float* Akey      = ws;                      // 1968*16*128 = 4,030,464
  float* Rkey      = Akey + 4030464;          // 116*16*128  =   237,568
  float* qsum      = Rkey + 237568;           // 1,024
  float* epsv      = qsum + 1024;             // 128
  float* comp_feat = epsv + 128;              // 20000*128   = 2,560,000
  float* wbe       = comp_feat + 2560000;     // 8*20000     =   160,000
  float* atts      = wbe + 160000;            // 160,000
  float* xbuf      = atts + 160000;           // 8*1968*128  = 2,015,232
  float* Qm        = xbuf + 2015232;
  float* Km        = Qm + 2015232;
  float* Vm        = Km + 2015232;
  float* attended  = Vm + 2015232;
  float* neigh     = attended + 2015232;      // zero group starts here
  float* importance= neigh + 2015232;         // 15,744
  unsigned* m1bits = (unsigned*)(importance + 15744);
  unsigned* m2bits = m1bits + 15744;
  int* selidx      = (int*)(m2bits + 15744);  // 1,568

  // zero neigh + importance + m1 + m2 in one memset
  (void)hipMemsetAsync(neigh, 0, (size_t)(2015232 + 3 * 15744) * sizeof(float),
                       stream);

  // small precomputes
  qsum_kernel<<<4, 256, 0, stream>>>(q_emb, W_tok, b_tok, qsum);
  eps_kernel<<<1, 128, 0, stream>>>(rel_mask, epsv);

  // key projections (linearity: per-entity / per-relation instead of per-edge)
  // Akey = ent_emb @ W_key            M=31488 K=256
  gemm_wmma_f32<<<1968, 256, 0, stream>>>(ent_emb, W_key, nullptr, Akey,
                                          N_ENT * S_TOK, 2 * HDIM);
  // Rkey = rel_emb @ W_key + b_key    M=1856
  gemm_wmma_f32<<<116, 256, 0, stream>>>(rel_emb, W_key, b_key, Rkey,
                                         NREL2 * S_TOK, 2 * HDIM);

  // per-edge features + raw attention logits
  edge_kernel<<<EDGES, 128, 0, stream>>>(Akey, Rkey, qsum, epsv, W_comp, b_comp,
                                         src, dst, rel_id, rel_mask,
                                         comp_feat, wbe);
  softmax_kernel<<<BQ, 256, 0, stream>>>(wbe, atts);

  // top-2 gating via two segment-max passes
  const int segBlocks = (BQ * EDGES + 255) / 256;
  seg_m1<<<segBlocks, 256, 0, stream>>>(atts, src, m1bits);
  seg_m2<<<segBlocks, 256, 0, stream>>>(atts, src, m1bits, m2bits);

  // scatter messages, layernorm
  msg_kernel<<<EDGES, 128, 0, stream>>>(atts, comp_feat, src, dst, m2bits, neigh);
  ln_kernel<<<BQ * N_ENT, 128, 0, stream>>>(neigh, xbuf);

  // Q/K/V projections: M=15744 K=128
  gemm_wmma_f32<<<984, 256, 0, stream>>>(xbuf, Wq, bq, Qm, BQ * N_ENT, HDIM);
  gemm_wmma_f32<<<984, 256, 0, stream>>>(xbuf, Wk, bk, Km, BQ * N_ENT, HDIM);
  gemm_wmma_f32<<<984, 256, 0, stream>>>(xbuf, Wv, bv, Vm, BQ * N_ENT, HDIM);

  // self-attention (streaming softmax, WMMA QK^T and P@V) + importance
  attn_kernel<<<123, 256, 0, stream>>>(Qm, Km, Vm, attended, importance);

  // top-196 selection and final projected/tanh output
  topk_kernel<<<BQ, 256, 0, stream>>>(importance, selidx);
  out_kernel<<<98, 256, 0, stream>>>(attended, selidx, W_kg, b_kg, out);
}